// GATModel_50440095924822
// MI455X (gfx1250) — compile-verified
//
#include <hip/hip_runtime.h>
#include <cstdint>
#include <cstddef>

typedef __attribute__((ext_vector_type(2))) float v2f;
typedef __attribute__((ext_vector_type(8))) float v8f;
typedef __attribute__((ext_vector_type(4))) unsigned uv4;
typedef __attribute__((ext_vector_type(8))) int iv8;
typedef __attribute__((ext_vector_type(4))) int iv4;

#define NEG_SLOPE 0.2f

#if defined(__HIP_DEVICE_COMPILE__) && __has_builtin(__builtin_amdgcn_tensor_load_to_lds)
#define HAVE_TDM 1
#endif

__device__ __forceinline__ float leaky(float v) { return v > 0.f ? v : NEG_SLOPE * v; }

// Order-preserving map float -> uint so atomicMax(uint) == float max.
// All finite-value keys are > 0, so a zeroed buffer acts as -inf.
__device__ __forceinline__ unsigned enc_key(float f) {
  unsigned u = __float_as_uint(f);
  return (u & 0x80000000u) ? ~u : (u | 0x80000000u);
}
__device__ __forceinline__ float dec_key(unsigned k) {
  unsigned u = (k & 0x80000000u) ? (k ^ 0x80000000u) : ~k;
  return __uint_as_float(u);
}

// ---- CDNA5 async global->LDS copy (ASYNCcnt) ------------------------------
__device__ __forceinline__ void async_ld_b128(unsigned lds_byte_off, const void* gaddr) {
  asm volatile("global_load_async_to_lds_b128 %0, %1, off"
               :: "v"(lds_byte_off), "v"(gaddr)
               : "memory");
}
__device__ __forceinline__ void wait_async0() {
  asm volatile("s_wait_asynccnt 0x0" ::: "memory");
}

#if defined(HAVE_TDM)
// ---- CDNA5 Tensor Data Mover: 2D tile Global->LDS (TENSORcnt) -------------
// D# packing per CDNA5 ISA.
__device__ __forceinline__ void tdm_load_tile_2d(const void* gsrc, unsigned lds_off,
                                                 unsigned row_len, unsigned nrows,
                                                 unsigned tensor_rows, unsigned row_stride) {
  unsigned long long ga = (unsigned long long)(uintptr_t)gsrc;
  uv4 g0;
  g0[0] = 1u;                                   // count=1 (valid user descriptor)
  g0[1] = lds_off;                              // lds_addr (bytes)
  g0[2] = (unsigned)ga;                         // global_addr[31:0]
  g0[3] = (unsigned)((ga >> 32) & 0x01FFFFFFu)  // global_addr[56:32]
          | (2u << 30);                         // type=2 ("image")
  iv8 g1;
  g1[0] = (int)(2u << 16);                      // wg_mask=0, data_size=2 (4 bytes)
  g1[1] = (int)(row_len << 16);                 // tensor_dim0[15:0]  (bits 63:48)
  g1[2] = (int)((row_len >> 16) | (tensor_rows << 16));  // dim0 hi | dim1 lo
  g1[3] = (int)((tensor_rows >> 16) | (row_len << 16));  // dim1 hi | tile_dim0
  g1[4] = (int)(nrows & 0xFFFFu);               // tile_dim1 (tile_dim2 = 0)
  g1[5] = (int)row_stride;                      // tensor_dim0_stride[31:0]
  g1[6] = 0;                                    // stride hi | dim1_stride lo
  g1[7] = 0;
  iv4 z4 = {0, 0, 0, 0};
#if __clang_major__ >= 23
  iv8 z8 = {0, 0, 0, 0, 0, 0, 0, 0};
  __builtin_amdgcn_tensor_load_to_lds(g0, g1, z4, z4, z8, 0);
#else
  __builtin_amdgcn_tensor_load_to_lds(g0, g1, z4, z4, 0);
#endif
}
#endif

// ---------------- GEMM1: h = x[N,512] @ W1[512,128], fp32 WMMA 16x16x4 -----
// Block = 256 thr = 8 waves; block owns a 16-row x 128-col strip.
// K chunks of 32 double-buffered in LDS: A tile via async-LDS loads,
// B tile via the Tensor Data Mover (wave 0), WMMA operands from DS loads.
#define KC 32
__global__ __launch_bounds__(256) void gemm_x_w1(const float* __restrict__ x,
                                                 const float* __restrict__ W,
                                                 float* __restrict__ h,
                                                 int n_nodes) {
  __shared__ float Ab[2][16 * 36];   // stride 36: 16B-aligned rows, conflict-free
  __shared__ float Bb[2][KC * 128];

  const int tid   = threadIdx.x;
  const int lane  = tid & 31;
  const int nTile = tid >> 5;        // 0..7 column tile per wave
  const int mTile = blockIdx.x;

  auto issue = [&](int c, int b) {
    if (tid < 128) {                 // A tile: 16 rows x 32 floats = 128 x b128
      int r  = tid >> 3;
      int ch = tid & 7;
      int row = min(mTile * 16 + r, n_nodes - 1);
      async_ld_b128((unsigned)(uintptr_t)&Ab[b][r * 36 + ch * 4],
                    x + (size_t)row * 512 + c * KC + ch * 4);
    }
#if defined(HAVE_TDM)
    if (tid < 32) {                  // wave 0 drives the TDM for the B tile
      tdm_load_tile_2d(W + (size_t)c * KC * 128,
                       (unsigned)(uintptr_t)&Bb[b][0],
                       128u, (unsigned)KC, 512u, 128u);
    }
#else
    for (int i = tid; i < KC * 128 / 4; i += 256)
      async_ld_b128((unsigned)(uintptr_t)&Bb[b][i * 4],
                    W + (size_t)c * KC * 128 + i * 4);
#endif
  };

  issue(0, 0);

  const int col  = nTile * 16 + (lane & 15);
  const int mrow = lane & 15;
  const int ko   = (lane >> 4) << 1;   // 0 or 2: upper half-wave holds K+2
  v8f acc = {};

  for (int c = 0; c < 512 / KC; ++c) {
    const int b = c & 1;
    wait_async0();
#if defined(HAVE_TDM)
    __builtin_amdgcn_s_wait_tensorcnt(0);
#endif
    __syncthreads();
    if (c + 1 < 512 / KC) issue(c + 1, b ^ 1);
    const float* Ap = &Ab[b][mrow * 36 + ko];
    const float* Bp = &Bb[b][(size_t)ko * 128 + col];
#pragma unroll
    for (int k = 0; k < KC; k += 4) {
      v2f a, bb;
      a.x  = Ap[k];
      a.y  = Ap[k + 1];
      bb.x = Bp[k * 128];
      bb.y = Bp[k * 128 + 128];
      acc = __builtin_amdgcn_wmma_f32_16x16x4_f32(false, a, false, bb,
                                                  (short)0, acc, false, false);
    }
    __syncthreads();
  }

  const int rBase = mTile * 16 + ((lane >> 4) << 3);
#pragma unroll
  for (int r = 0; r < 8; ++r) {
    int row = rBase + r;
    if (row < n_nodes) h[(size_t)row * 128 + col] = acc[r];
  }
}

// ---------------- GEMM2: h2 = h1[N,128] @ W2[128,40], fp32 WMMA ------------
__global__ __launch_bounds__(256) void gemm_h1_w2(const float* __restrict__ h1,
                                                  const float* __restrict__ W2,
                                                  float* __restrict__ h2,
                                                  int n_nodes) {
  const int gwave = (int)((blockIdx.x * blockDim.x + threadIdx.x) >> 5);
  const int lane  = threadIdx.x & 31;
  const int mTiles = (n_nodes + 15) >> 4;
  const int mTile = gwave / 3;
  const int nTile = gwave % 3;
  if (mTile >= mTiles) return;
  const int rowA = min(mTile * 16 + (lane & 15), n_nodes - 1);
  const int col  = nTile * 16 + (lane & 15);
  const int ko   = (lane >> 4) << 1;
  const bool cok = (col < 40);
  const float* hr = h1 + (size_t)rowA * 128;
  v8f acc = {};
  for (int k = 0; k < 128; k += 4) {
    v2f a, b;
    a.x = hr[k + ko];
    a.y = hr[k + ko + 1];
    b.x = cok ? W2[(size_t)(k + ko) * 40 + col] : 0.f;
    b.y = cok ? W2[(size_t)(k + ko + 1) * 40 + col] : 0.f;
    acc = __builtin_amdgcn_wmma_f32_16x16x4_f32(false, a, false, b,
                                                (short)0, acc, false, false);
  }
  if (cok) {
    const int rBase = mTile * 16 + ((lane >> 4) << 3);
#pragma unroll
    for (int r = 0; r < 8; ++r) {
      int row = rBase + r;
      if (row < n_nodes) h2[(size_t)row * 40 + col] = acc[r];
    }
  }
}

// ------------- per-node attention dot products (layer 1: 4 heads x 32) -----
__global__ void attn1(const float* __restrict__ h, const float* __restrict__ aw_s,
                      const float* __restrict__ aw_d, float* __restrict__ as,
                      float* __restrict__ ad, int n_nodes) {
  int t = blockIdx.x * blockDim.x + threadIdx.x;
  if (t >= n_nodes * 4) return;
  int node = t >> 2, hd = t & 3;
  const float* hp = h + (size_t)node * 128 + hd * 32;
  const float* s1 = aw_s + hd * 32;
  const float* s2 = aw_d + hd * 32;
  float d1 = 0.f, d2 = 0.f;
#pragma unroll
  for (int c = 0; c < 32; ++c) { float v = hp[c]; d1 += v * s1[c]; d2 += v * s2[c]; }
  as[t] = d1; ad[t] = d2;
}

// ------------- per-node attention dot products (layer 2: 1 head x 40) ------
__global__ void attn2(const float* __restrict__ h2, const float* __restrict__ aw_s,
                      const float* __restrict__ aw_d, float* __restrict__ as,
                      float* __restrict__ ad, int n_nodes) {
  int node = blockIdx.x * blockDim.x + threadIdx.x;
  if (node >= n_nodes) return;
  const float* hp = h2 + (size_t)node * 40;
  float d1 = 0.f, d2 = 0.f;
#pragma unroll
  for (int c = 0; c < 40; ++c) { float v = hp[c]; d1 += v * aw_s[c]; d2 += v * aw_d[c]; }
  as[node] = d1; ad[node] = d2;
}

// ------------- edge pass A (layer 1): segment max via encoded atomicMax ----
__global__ void edge_max1(const int* __restrict__ src, const int* __restrict__ dst,
                          const float* __restrict__ as, const float* __restrict__ ad,
                          unsigned* __restrict__ m1, int n_edges, int n_nodes) {
  int e = blockIdx.x * blockDim.x + threadIdx.x;
  int tot = n_edges + n_nodes;
  if (e >= tot) return;
  int s = (e < n_edges) ? src[e] : (e - n_edges);   // self loops appended
  int d = (e < n_edges) ? dst[e] : (e - n_edges);
#pragma unroll
  for (int hd = 0; hd < 4; ++hd) {
    float v = leaky(as[s * 4 + hd] + ad[d * 4 + hd]);
    atomicMax(&m1[d * 4 + hd], enc_key(v));
  }
}

// ------------- edge pass B (layer 1): exp + weighted scatter ---------------
__global__ __launch_bounds__(256) void edge_agg1(
    const int* __restrict__ src, const int* __restrict__ dst,
    const float* __restrict__ as, const float* __restrict__ ad,
    const unsigned* __restrict__ m1, const float* __restrict__ hfeat,
    float* __restrict__ denom, float* __restrict__ agg,
    int n_edges, int n_nodes) {
  int e = (int)((blockIdx.x * blockDim.x + threadIdx.x) >> 5);
  int lane = threadIdx.x & 31;
  int tot = n_edges + n_nodes;
  if (e >= tot) return;
  int s = (e < n_edges) ? src[e] : (e - n_edges);
  int d = (e < n_edges) ? dst[e] : (e - n_edges);
  const float* hsrc = hfeat + (size_t)s * 128;
  __builtin_prefetch(hsrc + lane * 4, 0, 0);   // global_prefetch_b8
  float exh[4];
#pragma unroll
  for (int hd = 0; hd < 4; ++hd) {
    float v = leaky(as[s * 4 + hd] + ad[d * 4 + hd]);
    exh[hd] = __expf(v - dec_key(m1[d * 4 + hd]));
  }
  if (lane < 4) atomicAdd(&denom[d * 4 + lane], exh[lane]);
  const float4 hv = *(const float4*)(hsrc + lane * 4);
  float* ao = agg + (size_t)d * 128 + lane * 4;
  float ex = exh[lane >> 3];
  atomicAdd(ao + 0, ex * hv.x);
  atomicAdd(ao + 1, ex * hv.y);
  atomicAdd(ao + 2, ex * hv.z);
  atomicAdd(ao + 3, ex * hv.w);
}

// ------------- layer-1 finish: normalize, +bias, ELU -----------------------
__global__ void node_fin1(const float* __restrict__ agg, const float* __restrict__ denom,
                          const float* __restrict__ b1, float* __restrict__ h1,
                          int n_nodes) {
  int t = blockIdx.x * blockDim.x + threadIdx.x;
  if (t >= n_nodes * 128) return;
  int node = t >> 7, f = t & 127, hd = f >> 5;
  float v = agg[t] / (denom[node * 4 + hd] + 1e-9f) + b1[f];
  h1[t] = v > 0.f ? v : (__expf(v) - 1.f);
}

// ------------- edge pass A (layer 2) ---------------------------------------
__global__ void edge_max2(const int* __restrict__ src, const int* __restrict__ dst,
                          const float* __restrict__ as, const float* __restrict__ ad,
                          unsigned* __restrict__ m2, int n_edges, int n_nodes) {
  int e = blockIdx.x * blockDim.x + threadIdx.x;
  int tot = n_edges + n_nodes;
  if (e >= tot) return;
  int s = (e < n_edges) ? src[e] : (e - n_edges);
  int d = (e < n_edges) ? dst[e] : (e - n_edges);
  float v = leaky(as[s] + ad[d]);
  atomicMax(&m2[d], enc_key(v));
}

// ------------- edge pass B (layer 2): 40-wide scatter ----------------------
__global__ __launch_bounds__(256) void edge_agg2(
    const int* __restrict__ src, const int* __restrict__ dst,
    const float* __restrict__ as, const float* __restrict__ ad,
    const unsigned* __restrict__ m2, const float* __restrict__ h2lin,
    float* __restrict__ denom, float* __restrict__ agg,
    int n_edges, int n_nodes) {
  int e = (int)((blockIdx.x * blockDim.x + threadIdx.x) >> 5);
  int lane = threadIdx.x & 31;
  int tot = n_edges + n_nodes;
  if (e >= tot) return;
  int s = (e < n_edges) ? src[e] : (e - n_edges);
  int d = (e < n_edges) ? dst[e] : (e - n_edges);
  const float* hs = h2lin + (size_t)s * 40;
  __builtin_prefetch(hs + lane, 0, 0);
  float v = leaky(as[s] + ad[d]);
  float ex = __expf(v - dec_key(m2[d]));
  if (lane == 0) atomicAdd(&denom[d], ex);
  float* ao = agg + (size_t)d * 40;
  atomicAdd(&ao[lane], ex * hs[lane]);
  if (lane < 8) atomicAdd(&ao[lane + 32], ex * hs[lane + 32]);
}

// ------------- layer-2 finish: normalize, +bias, softmax -------------------
__global__ void node_fin2(const float* __restrict__ agg, const float* __restrict__ denom,
                          const float* __restrict__ b2, float* __restrict__ out,
                          int n_nodes) {
  int node = blockIdx.x * blockDim.x + threadIdx.x;
  if (node >= n_nodes) return;
  float inv = 1.f / (denom[node] + 1e-9f);
  float vals[40];
  float mx = -3.4e38f;
#pragma unroll
  for (int c = 0; c < 40; ++c) {
    float v = agg[(size_t)node * 40 + c] * inv + b2[c];
    vals[c] = v;
    mx = fmaxf(mx, v);
  }
  float sum = 0.f;
#pragma unroll
  for (int c = 0; c < 40; ++c) { float ev = __expf(vals[c] - mx); vals[c] = ev; sum += ev; }
  float is = 1.f / sum;
#pragma unroll
  for (int c = 0; c < 40; ++c) out[(size_t)node * 40 + c] = vals[c] * is;
}

extern "C" void kernel_launch(void* const* d_in, const int* in_sizes, int n_in,
                              void* d_out, int out_size, void* d_ws, size_t ws_size,
                              hipStream_t stream) {
  const float* x    = (const float*)d_in[0];
  const int*   src  = (const int*)d_in[1];
  const int*   dst  = (const int*)d_in[2];
  const float* W1   = (const float*)d_in[3];
  const float* aw1s = (const float*)d_in[4];
  const float* aw1d = (const float*)d_in[5];
  const float* b1   = (const float*)d_in[6];
  const float* W2   = (const float*)d_in[7];
  const float* aw2s = (const float*)d_in[8];
  const float* aw2d = (const float*)d_in[9];
  const float* b2   = (const float*)d_in[10];
  float* out = (float*)d_out;

  const int n_nodes = in_sizes[0] / 512;
  const int n_edges = in_sizes[1];
  const int tot_e   = n_edges + n_nodes;

  float* p = (float*)d_ws;
  float*    hfeat = p; p += (size_t)n_nodes * 128;   // x@W1, later h1 (post-ELU)
  float*    agg1  = p; p += (size_t)n_nodes * 128;
  float*    as1   = p; p += (size_t)n_nodes * 4;
  float*    ad1   = p; p += (size_t)n_nodes * 4;
  unsigned* m1    = (unsigned*)p; p += (size_t)n_nodes * 4;
  float*    den1  = p; p += (size_t)n_nodes * 4;
  float*    h2lin = p; p += (size_t)n_nodes * 40;
  float*    as2   = p; p += (size_t)n_nodes;
  float*    ad2   = p; p += (size_t)n_nodes;
  unsigned* m2    = (unsigned*)p; p += (size_t)n_nodes;
  float*    den2  = p; p += (size_t)n_nodes;
  float*    agg2  = p; p += (size_t)n_nodes * 40;

  hipMemsetAsync(agg1, 0, (size_t)n_nodes * 128 * 4, stream);
  hipMemsetAsync(m1,   0, (size_t)n_nodes * 4 * 4, stream);
  hipMemsetAsync(den1, 0, (size_t)n_nodes * 4 * 4, stream);
  hipMemsetAsync(agg2, 0, (size_t)n_nodes * 40 * 4, stream);
  hipMemsetAsync(m2,   0, (size_t)n_nodes * 4, stream);
  hipMemsetAsync(den2, 0, (size_t)n_nodes * 4, stream);

  const int mTiles = (n_nodes + 15) / 16;

  gemm_x_w1<<<mTiles, 256, 0, stream>>>(x, W1, hfeat, n_nodes);
  attn1<<<(n_nodes * 4 + 255) / 256, 256, 0, stream>>>(hfeat, aw1s, aw1d, as1, ad1, n_nodes);
  edge_max1<<<(tot_e + 255) / 256, 256, 0, stream>>>(src, dst, as1, ad1, m1, n_edges, n_nodes);
  edge_agg1<<<(int)(((size_t)tot_e * 32 + 255) / 256), 256, 0, stream>>>(
      src, dst, as1, ad1, m1, hfeat, den1, agg1, n_edges, n_nodes);
  node_fin1<<<(int)(((size_t)n_nodes * 128 + 255) / 256), 256, 0, stream>>>(
      agg1, den1, b1, hfeat, n_nodes);   // h1 overwrites hfeat

  const int waves2 = mTiles * 3;
  gemm_h1_w2<<<(int)(((size_t)waves2 * 32 + 255) / 256), 256, 0, stream>>>(hfeat, W2, h2lin, n_nodes);
  attn2<<<(n_nodes + 255) / 256, 256, 0, stream>>>(h2lin, aw2s, aw2d, as2, ad2, n_nodes);
  edge_max2<<<(tot_e + 255) / 256, 256, 0, stream>>>(src, dst, as2, ad2, m2, n_edges, n_nodes);
  edge_agg2<<<(int)(((size_t)tot_e * 32 + 255) / 256), 256, 0, stream>>>(
      src, dst, as2, ad2, m2, h2lin, den2, agg2, n_edges, n_nodes);
  node_fin2<<<(n_nodes + 255) / 256, 256, 0, stream>>>(agg2, den2, b2, out, n_nodes);
}